// CrossLayerLight_78426102825593
// MI455X (gfx1250) — compile-verified
//
#include <hip/hip_runtime.h>

// ---------------- problem constants ----------------
#define NPTS   4096
#define BATCH  2
#define CH     128
#define CINF   64
#define KNN_K  16
#define GRP    8
#define CPG    16            // channels per group
#define COLS   (NPTS*KNN_K)  // 65536 columns per batch in the [C, N*K] view
#define EPSV   1e-5f

typedef __attribute__((ext_vector_type(2))) float v2f;
typedef __attribute__((ext_vector_type(8))) float v8f;

// =====================================================================
// KNN: brute-force top-16 nearest (unsorted set; downstream ops over K
// are permutation invariant: GN reduces over all of (C/G,N,K), maxpool
// over K). LDS-tiled reference points, running-max replacement.
// =====================================================================
__global__ __launch_bounds__(256) void knn_kernel(const float* __restrict__ q,
                                                  const float* __restrict__ r,
                                                  int* __restrict__ idx) {
    __shared__ float sx[256], sy[256], sz[256];
    int gid = blockIdx.x * 256 + threadIdx.x;
    int b = gid / NPTS, i = gid % NPTS;
    const float* qb = q + (size_t)b * 3 * NPTS;
    const float* rb = r + (size_t)b * 3 * NPTS;
    float qx = qb[i], qy = qb[NPTS + i], qz = qb[2 * NPTS + i];

    float bd[KNN_K];
    int   bj[KNN_K];
#pragma unroll
    for (int s = 0; s < KNN_K; ++s) { bd[s] = 3.4e38f; bj[s] = 0; }
    float worst = 3.4e38f;
    int   wslot = 0;

    for (int t = 0; t < NPTS; t += 256) {
        __syncthreads();
        sx[threadIdx.x] = rb[t + threadIdx.x];
        sy[threadIdx.x] = rb[NPTS + t + threadIdx.x];
        sz[threadIdx.x] = rb[2 * NPTS + t + threadIdx.x];
        __syncthreads();
        for (int jj = 0; jj < 256; ++jj) {
            float dx = qx - sx[jj], dy = qy - sy[jj], dz = qz - sz[jj];
            float d = dx * dx + dy * dy + dz * dz;
            if (d < worst) {
                int j = t + jj;
#pragma unroll
                for (int s = 0; s < KNN_K; ++s)
                    if (s == wslot) { bd[s] = d; bj[s] = j; }
                worst = bd[0]; wslot = 0;
#pragma unroll
                for (int s = 1; s < KNN_K; ++s)
                    if (bd[s] > worst) { worst = bd[s]; wslot = s; }
            }
        }
    }
    int* ob = idx + ((size_t)b * NPTS + i) * KNN_K;
#pragma unroll
    for (int s = 0; s < KNN_K; ++s) ob[s] = bj[s];
}

// =====================================================================
// 1x1 Conv1d as GEMM: Y[b,o,n] = sum_c W[o,c] X[b,c,n] + bias[o]
// One wave -> one 16x16 (oc x n) tile via V_WMMA_F32_16X16X4_F32.
// A (16x4 f32): lane<16 holds row m=lane, K pair {0,1}; lane>=16 K {2,3}.
// B (4x16 f32): mirrored (column per lane, K pair split across halves).
// =====================================================================
__global__ __launch_bounds__(256) void conv1d_wmma_kernel(const float* __restrict__ W,
                                                          const float* __restrict__ bias,
                                                          const float* __restrict__ X,
                                                          float* __restrict__ Y, int cin) {
    int wave = (blockIdx.x * 256 + threadIdx.x) >> 5;
    int lane = threadIdx.x & 31;
    const int ntiles = NPTS / 16;                 // 256
    int b   = wave / (8 * ntiles);
    int rem = wave % (8 * ntiles);
    int oc0 = (rem / ntiles) * 16;
    int n0  = (rem % ntiles) * 16;
    int m   = lane & 15;
    int kh  = (lane >> 4) * 2;

    const float* Xb = X + (size_t)b * cin * NPTS;
    v8f acc = {};
    for (int c = 0; c < cin; c += 4) {
        v2f a, bb;
        a.x  = W[(size_t)(oc0 + m) * cin + c + kh];
        a.y  = W[(size_t)(oc0 + m) * cin + c + kh + 1];
        bb.x = Xb[(size_t)(c + kh) * NPTS + n0 + m];
        bb.y = Xb[(size_t)(c + kh + 1) * NPTS + n0 + m];
        acc = __builtin_amdgcn_wmma_f32_16x16x4_f32(false, a, false, bb,
                                                    (short)0, acc, false, false);
    }
    float* Yb = Y + (size_t)b * CH * NPTS;
    int n  = n0 + m;
    int mb = (lane >> 4) * 8;
#pragma unroll
    for (int r = 0; r < 8; ++r) {
        int o = oc0 + mb + r;
        Yb[(size_t)o * NPTS + n] = acc[r] + bias[o];
    }
}

// =====================================================================
// cross stage 1: pre[b,c,n,k] = P2[b,c,idx] + P1[b,c,n] + posW[c,:].dir
//                + posB[c]; accumulate GN sum/sumsq per (b,group).
// Block = 256 threads = one (b, 16-point tile); thread = one (n,k) column.
// =====================================================================
__global__ __launch_bounds__(256) void cross_pre_kernel(const int* __restrict__ idx,
                                                        const float* __restrict__ qxyz,
                                                        const float* __restrict__ rxyz,
                                                        const float* __restrict__ P1,
                                                        const float* __restrict__ P2,
                                                        const float* __restrict__ posW,
                                                        const float* __restrict__ posB,
                                                        float* __restrict__ pre,
                                                        float* __restrict__ sums) {
    __shared__ float ssum[GRP], ssq[GRP];
    int b  = blockIdx.x / (NPTS / 16);
    int nt = blockIdx.x % (NPTS / 16);
    int nl = threadIdx.x >> 4;
    int k  = threadIdx.x & 15;
    int n  = nt * 16 + nl;
    int j  = idx[((size_t)b * NPTS + n) * KNN_K + k];

    const float* qb = qxyz + (size_t)b * 3 * NPTS;
    const float* rb = rxyz + (size_t)b * 3 * NPTS;
    float dx = rb[j] - qb[n];
    float dy = rb[NPTS + j] - qb[NPTS + n];
    float dz = rb[2 * NPTS + j] - qb[2 * NPTS + n];

    if (threadIdx.x < GRP) { ssum[threadIdx.x] = 0.f; ssq[threadIdx.x] = 0.f; }
    __syncthreads();

    const float* P1b = P1 + (size_t)b * CH * NPTS;
    const float* P2b = P2 + (size_t)b * CH * NPTS;
    float* preb = pre + (size_t)b * CH * COLS;
    size_t col = (size_t)n * KNN_K + k;
    int lane = threadIdx.x & 31;

    for (int g = 0; g < GRP; ++g) {
        float s = 0.f, s2 = 0.f;
#pragma unroll
        for (int ci = 0; ci < CPG; ++ci) {
            int c = g * CPG + ci;
            float v = P2b[(size_t)c * NPTS + j] + P1b[(size_t)c * NPTS + n]
                    + posW[c * 3 + 0] * dx + posW[c * 3 + 1] * dy
                    + posW[c * 3 + 2] * dz + posB[c];
            preb[(size_t)c * COLS + col] = v;
            s += v; s2 += v * v;
        }
#pragma unroll
        for (int off = 16; off; off >>= 1) { s += __shfl_xor(s, off); s2 += __shfl_xor(s2, off); }
        if (lane == 0) { atomicAdd(&ssum[g], s); atomicAdd(&ssq[g], s2); }
    }
    __syncthreads();
    if (threadIdx.x < GRP) {
        atomicAdd(&sums[(b * GRP + threadIdx.x) * 2 + 0], ssum[threadIdx.x]);
        atomicAdd(&sums[(b * GRP + threadIdx.x) * 2 + 1], ssq[threadIdx.x]);
    }
}

// =====================================================================
// GN stats -> mean + rsqrt(var+eps)
// =====================================================================
__global__ void gn_finalize_kernel(const float* __restrict__ sums,
                                   float* __restrict__ mi, float count) {
    int t = threadIdx.x;
    if (t < BATCH * GRP) {
        float s = sums[t * 2], s2 = sums[t * 2 + 1];
        float mu  = s / count;
        float var = s2 / count - mu * mu;
        mi[t * 2]     = mu;
        mi[t * 2 + 1] = rsqrtf(var + EPSV);
    }
}

// =====================================================================
// Heavy kernel: x = lrelu(gn(pre_in)) staged in LDS (128x32 tile shared
// by 8 waves), then out = Wm(128x128) . x + bm via f32 WMMA; fused
// accumulation of NEXT GN's stats (out-channel block of 16 == 1 group).
// =====================================================================
__global__ __launch_bounds__(256) void gn_conv_kernel(const float* __restrict__ pre_in,
                                                      const float* __restrict__ mi,
                                                      const float* __restrict__ gamma,
                                                      const float* __restrict__ beta,
                                                      const float* __restrict__ Wm,
                                                      const float* __restrict__ bm,
                                                      float* __restrict__ pre_out,
                                                      float* __restrict__ sums) {
    __shared__ float xs[CH * 32];
    __shared__ float ssum[GRP], ssq[GRP];
    int b  = blockIdx.x / (COLS / 32);
    int ct = blockIdx.x % (COLS / 32);
    size_t col0 = (size_t)ct * 32;
    const float* pin = pre_in + (size_t)b * CH * COLS;
    int tid = threadIdx.x;

#pragma unroll
    for (int it = 0; it < 16; ++it) {
        int e  = it * 256 + tid;     // 0..4095 over 128x32 tile
        int c  = e >> 5;
        int cc = e & 31;
        int g  = c >> 4;
        float mu  = mi[(b * GRP + g) * 2];
        float inv = mi[(b * GRP + g) * 2 + 1];
        float v = pin[(size_t)c * COLS + col0 + cc];
        v = (v - mu) * inv * gamma[c] + beta[c];
        v = v >= 0.f ? v : 0.1f * v;
        xs[c * 32 + cc] = v;
    }
    __syncthreads();

    int wave = tid >> 5, lane = tid & 31;
    int oc0 = wave * 16;             // == one GN group of the output
    int m   = lane & 15;
    int kh  = (lane >> 4) * 2;
    v8f acc0 = {}, acc1 = {};
    for (int c = 0; c < CH; c += 4) {
        v2f a, b0, b1;
        a.x  = Wm[(size_t)(oc0 + m) * CH + c + kh];
        a.y  = Wm[(size_t)(oc0 + m) * CH + c + kh + 1];
        b0.x = xs[(c + kh) * 32 + m];      b0.y = xs[(c + kh + 1) * 32 + m];
        b1.x = xs[(c + kh) * 32 + 16 + m]; b1.y = xs[(c + kh + 1) * 32 + 16 + m];
        acc0 = __builtin_amdgcn_wmma_f32_16x16x4_f32(false, a, false, b0,
                                                     (short)0, acc0, false, false);
        acc1 = __builtin_amdgcn_wmma_f32_16x16x4_f32(false, a, false, b1,
                                                     (short)0, acc1, false, false);
    }

    float* pout = pre_out + (size_t)b * CH * COLS;
    int mb = (lane >> 4) * 8;
    float s = 0.f, s2 = 0.f;
#pragma unroll
    for (int r = 0; r < 8; ++r) {
        int o = oc0 + mb + r;
        float v0 = acc0[r] + bm[o];
        float v1 = acc1[r] + bm[o];
        pout[(size_t)o * COLS + col0 + m]      = v0;
        pout[(size_t)o * COLS + col0 + 16 + m] = v1;
        s += v0 + v1; s2 += v0 * v0 + v1 * v1;
    }
#pragma unroll
    for (int off = 16; off; off >>= 1) { s += __shfl_xor(s, off); s2 += __shfl_xor(s2, off); }
    if (lane == 0) { ssum[wave] = s; ssq[wave] = s2; }   // one wave per group slot
    __syncthreads();
    if (tid < GRP) {
        atomicAdd(&sums[(b * GRP + tid) * 2 + 0], ssum[tid]);
        atomicAdd(&sums[(b * GRP + tid) * 2 + 1], ssq[tid]);
    }
}

// =====================================================================
// Final stage of a cross: lrelu(gn(pre)) then max over K -> [B,C,N]
// =====================================================================
__global__ __launch_bounds__(256) void gn_max_kernel(const float* __restrict__ pre,
                                                     const float* __restrict__ mi,
                                                     const float* __restrict__ gamma,
                                                     const float* __restrict__ beta,
                                                     float* __restrict__ out) {
    int gid = blockIdx.x * 256 + threadIdx.x;          // [B,C,N] flat
    int b   = gid / (CH * NPTS);
    int rem = gid % (CH * NPTS);
    int c   = rem / NPTS;
    int g   = c >> 4;
    float mu  = mi[(b * GRP + g) * 2];
    float inv = mi[(b * GRP + g) * 2 + 1];
    float ga = gamma[c] * inv, be = beta[c];
    const float* p = pre + (size_t)gid * KNN_K;
    float mx = -3.4e38f;
#pragma unroll
    for (int k = 0; k < KNN_K; ++k) {
        float v = (p[k] - mu) * ga + be;
        v = v >= 0.f ? v : 0.1f * v;
        mx = fmaxf(mx, v);
    }
    out[gid] = mx;
}

// =====================================================================
// Host-side helper: one full _cross() stage
// =====================================================================
static void run_cross(hipStream_t stream, const int* idx,
                      const float* qx, const float* rx,
                      const float* P1, const float* P2,
                      const float* posW, const float* posB,
                      const float* gnG, const float* gnB,
                      const float* const* mlpW, const float* const* mlpB,
                      const float* const* mlpG, const float* const* mlpBt, int nlayers,
                      float* preA, float* preB, float* stats, float* out) {
    float* sums = stats;
    float* mi   = stats + 32;
    const float cnt = (float)((size_t)CPG * NPTS * KNN_K);

    hipMemsetAsync(sums, 0, 32 * sizeof(float), stream);
    cross_pre_kernel<<<BATCH * (NPTS / 16), 256, 0, stream>>>(idx, qx, rx, P1, P2,
                                                              posW, posB, preA, sums);
    gn_finalize_kernel<<<1, 32, 0, stream>>>(sums, mi, cnt);

    const float* curG = gnG;
    const float* curB = gnB;
    float* pin  = preA;
    float* pout = preB;
    for (int l = 0; l < nlayers; ++l) {
        hipMemsetAsync(sums, 0, 32 * sizeof(float), stream);
        gn_conv_kernel<<<BATCH * (COLS / 32), 256, 0, stream>>>(pin, mi, curG, curB,
                                                                mlpW[l], mlpB[l], pout, sums);
        gn_finalize_kernel<<<1, 32, 0, stream>>>(sums, mi, cnt);
        curG = mlpG[l]; curB = mlpBt[l];
        float* t = pin; pin = pout; pout = t;
    }
    gn_max_kernel<<<(BATCH * CH * NPTS) / 256, 256, 0, stream>>>(pin, mi, curG, curB, out);
}

// =====================================================================
extern "C" void kernel_launch(void* const* d_in, const int* in_sizes, int n_in,
                              void* d_out, int out_size, void* d_ws, size_t ws_size,
                              hipStream_t stream) {
    (void)in_sizes; (void)n_in; (void)out_size; (void)ws_size;
    const float* pc1     = (const float*)d_in[0];
    const float* pc2     = (const float*)d_in[1];
    const float* feat1   = (const float*)d_in[2];
    const float* feat2   = (const float*)d_in[3];
    const float* t11_w   = (const float*)d_in[4];
    const float* t11_b   = (const float*)d_in[5];
    const float* t22_w   = (const float*)d_in[6];
    const float* t22_b   = (const float*)d_in[7];
    const float* pos1_w  = (const float*)d_in[8];
    const float* pos1_b  = (const float*)d_in[9];
    const float* gn1_g   = (const float*)d_in[10];
    const float* gn1_b   = (const float*)d_in[11];
    const float* m1a_w   = (const float*)d_in[12];
    const float* m1a_b   = (const float*)d_in[13];
    const float* m1a_g   = (const float*)d_in[14];
    const float* m1a_bt  = (const float*)d_in[15];
    const float* m1b_w   = (const float*)d_in[16];
    const float* m1b_b   = (const float*)d_in[17];
    const float* m1b_g   = (const float*)d_in[18];
    const float* m1b_bt  = (const float*)d_in[19];
    const float* t1_w    = (const float*)d_in[20];
    const float* t1_b    = (const float*)d_in[21];
    const float* t2_w    = (const float*)d_in[22];
    const float* t2_b    = (const float*)d_in[23];
    const float* pos2_w  = (const float*)d_in[24];
    const float* pos2_b  = (const float*)d_in[25];
    const float* gn2_g   = (const float*)d_in[26];
    const float* gn2_b   = (const float*)d_in[27];
    const float* m2a_w   = (const float*)d_in[28];
    const float* m2a_b   = (const float*)d_in[29];
    const float* m2a_g   = (const float*)d_in[30];
    const float* m2a_bt  = (const float*)d_in[31];

    // ---- workspace layout (floats) ----
    float* ws = (float*)d_ws;
    int*   idx12 = (int*)ws;                       // B*N*K = 131072
    int*   idx21 = (int*)(ws + 131072);            // 131072
    float* lf1   = ws + 262144;                    // each B*C*N = 1048576
    float* lf2   = lf1 + 1048576;
    float* lg1   = lf2 + 1048576;
    float* lg2   = lg1 + 1048576;
    float* xout  = lg2 + 1048576;                  // cross output (reused)
    float* preA  = xout + 1048576;                 // B*C*N*K = 16777216
    float* preB  = preA + 16777216;
    float* stats = preB + 16777216;                // 64 floats

    float* feat1_new   = (float*)d_out;                       // [2,128,4096]
    float* feat2_new   = feat1_new + BATCH * CH * NPTS;
    float* feat1_final = feat2_new + BATCH * CH * NPTS;

    // ---- KNN both directions ----
    knn_kernel<<<(BATCH * NPTS) / 256, 256, 0, stream>>>(pc1, pc2, idx12);
    knn_kernel<<<(BATCH * NPTS) / 256, 256, 0, stream>>>(pc2, pc1, idx21);

    // ---- linear lifts (WMMA GEMMs, Cin=64) ----
    const int c1dBlocks = (BATCH * 8 * (NPTS / 16) * 32) / 256;  // 512
    conv1d_wmma_kernel<<<c1dBlocks, 256, 0, stream>>>(t11_w, t11_b, feat1, lf1, CINF);
    conv1d_wmma_kernel<<<c1dBlocks, 256, 0, stream>>>(t22_w, t22_b, feat2, lf2, CINF);
    conv1d_wmma_kernel<<<c1dBlocks, 256, 0, stream>>>(t11_w, t11_b, feat2, lg1, CINF);
    conv1d_wmma_kernel<<<c1dBlocks, 256, 0, stream>>>(t22_w, t22_b, feat1, lg2, CINF);

    const float* mlp1W[2]  = { m1a_w, m1b_w };
    const float* mlp1B[2]  = { m1a_b, m1b_b };
    const float* mlp1G[2]  = { m1a_g, m1b_g };
    const float* mlp1Bt[2] = { m1a_bt, m1b_bt };

    // ---- cross 1: pc1 queries, pc2 neighbors ----
    run_cross(stream, idx12, pc1, pc2, lf1, lf2, pos1_w, pos1_b, gn1_g, gn1_b,
              mlp1W, mlp1B, mlp1G, mlp1Bt, 2, preA, preB, stats, xout);
    conv1d_wmma_kernel<<<c1dBlocks, 256, 0, stream>>>(t1_w, t1_b, xout, feat1_new, CH);

    // ---- cross 2: pc2 queries, pc1 neighbors ----
    run_cross(stream, idx21, pc2, pc1, lg1, lg2, pos1_w, pos1_b, gn1_g, gn1_b,
              mlp1W, mlp1B, mlp1G, mlp1Bt, 2, preA, preB, stats, xout);
    conv1d_wmma_kernel<<<c1dBlocks, 256, 0, stream>>>(t2_w, t2_b, xout, feat2_new, CH);

    // ---- cross 3: pc1 queries, pc2 neighbors, new features ----
    const float* mlp2W[1]  = { m2a_w };
    const float* mlp2B[1]  = { m2a_b };
    const float* mlp2G[1]  = { m2a_g };
    const float* mlp2Bt[1] = { m2a_bt };
    run_cross(stream, idx12, pc1, pc2, feat1_new, feat2_new, pos2_w, pos2_b, gn2_g, gn2_b,
              mlp2W, mlp2B, mlp2G, mlp2Bt, 1, preA, preB, stats, feat1_final);
}